// DQN_71554155152369
// MI455X (gfx1250) — compile-verified
//
#include <hip/hip_runtime.h>
#include <hip/hip_bf16.h>

typedef __attribute__((ext_vector_type(2))) float v2f;
typedef __attribute__((ext_vector_type(8))) float v8f;

// ---------------------------------------------------------------------------
// Kernel 1: per-board feature extraction.
// Board: 32x32, channel 0 = board state, channel 1 = prev-head marker.
// Output per board: 8 floats in d_ws: [free0, free1, free2, rotx, roty, 0,0,0]
// (stride 8 so the MLP kernel can treat K as padded-to-8 for WMMA chunks).
// ---------------------------------------------------------------------------
__global__ void snake_feat_kernel(const float* __restrict__ x,
                                  float* __restrict__ feat) {
  const int b   = blockIdx.x;
  const int tid = threadIdx.x;              // 256 threads, 4 cells each
  const float* __restrict__ g0 = x + (size_t)b * 3072;
  const float* __restrict__ g1 = g0 + 1024;

  __shared__ float    s_g0[1024];
  __shared__ unsigned s_min[3];             // min flat index: food, head, prev
  __shared__ int      s_head[2];
  __shared__ int      s_dirs[3][2];
  __shared__ int      s_rot[2];
  __shared__ unsigned char s_freev[3][32];
  __shared__ unsigned char s_inb[3][32];

  if (tid < 3) s_min[tid] = 0xFFFFFFFFu;
  __syncthreads();

  // --- stage channel 0 into LDS, build min-index keys (argmax-of-mask) ---
  const int i4 = tid * 4;
  const float4 v0 = *reinterpret_cast<const float4*>(g0 + i4);
  const float4 v1 = *reinterpret_cast<const float4*>(g1 + i4);
  *reinterpret_cast<float4*>(&s_g0[i4]) = v0;

  unsigned fkey = 0xFFFFFFFFu, hkey = 0xFFFFFFFFu, pkey = 0xFFFFFFFFu;
  const float a0[4] = {v0.x, v0.y, v0.z, v0.w};
  const float a1[4] = {v1.x, v1.y, v1.z, v1.w};
#pragma unroll
  for (int j = 0; j < 4; ++j) {
    const unsigned idx = (unsigned)(i4 + j);
    if (a0[j] == -1.0f) fkey = min(fkey, idx);
    if (a0[j] ==  1.0f) hkey = min(hkey, idx);
    if (a1[j] ==  1.0f) pkey = min(pkey, idx);
  }
  if (fkey != 0xFFFFFFFFu) atomicMin(&s_min[0], fkey);
  if (hkey != 0xFFFFFFFFu) atomicMin(&s_min[1], hkey);
  if (pkey != 0xFFFFFFFFu) atomicMin(&s_min[2], pkey);
  __syncthreads();

  // --- direction / rotation math (one thread) ---
  if (tid == 0) {
    unsigned fi = s_min[0]; if (fi == 0xFFFFFFFFu) fi = 0;  // argmax(all-F)=0
    unsigned hi = s_min[1]; if (hi == 0xFFFFFFFFu) hi = 0;
    unsigned pi = s_min[2]; if (pi == 0xFFFFFFFFu) pi = 0;
    const int fx = (int)(fi >> 5), fy = (int)(fi & 31);
    const int hx = (int)(hi >> 5), hy = (int)(hi & 31);
    const int px = (int)(pi >> 5), py = (int)(pi & 31);
    const int dx = hx - px, dy = hy - py;
    const int ddx = fx - hx, ddy = fy - hy;
    const int k = (dx == 1) ? 0 : (dy == -1) ? 1 : (dx == -1) ? 2 : 3;
    int rx, ry;
    switch (k) {
      case 0:  rx =  ddx; ry =  ddy; break;
      case 1:  rx = -ddy; ry =  ddx; break;
      case 2:  rx = -ddx; ry = -ddy; break;
      default: rx =  ddy; ry = -ddx; break;
    }
    s_rot[0] = rx; s_rot[1] = ry;
    s_head[0] = hx; s_head[1] = hy;
    s_dirs[0][0] = -dy; s_dirs[0][1] =  dx;   // left
    s_dirs[1][0] =  dx; s_dirs[1][1] =  dy;   // straight
    s_dirs[2][0] =  dy; s_dirs[2][1] = -dx;   // right
  }
  __syncthreads();

  // --- parallel ray-march: 3 directions x 32 steps ---
  if (tid < 96) {
    const int dir = tid >> 5;
    const int i   = tid & 31;       // step index (m = i+1)
    const int m   = i + 1;
    const int cx = s_head[0] + m * s_dirs[dir][0];
    const int cy = s_head[1] + m * s_dirs[dir][1];
    const bool inb = (cx >= 0) & (cx < 32) & (cy >= 0) & (cy < 32);
    const int ci = min(max(cx, 0), 31);
    const int cj = min(max(cy, 0), 31);
    const float val = s_g0[ci * 32 + cj];
    s_freev[dir][i] = (unsigned char)((val == 0.0f) | (val == -1.0f));
    s_inb[dir][i]   = (unsigned char)inb;
  }
  __syncthreads();

  // --- first step where (freev[i-1] & inb[i]) fails; default 0 ---
  if (tid < 3) {
    int freeLen = 0;
    for (int i = 0; i < 32; ++i) {
      const unsigned char fvp = s_freev[tid][(i == 0) ? 0 : i - 1];
      if (!(fvp & s_inb[tid][i])) { freeLen = i; break; }
    }
    feat[(size_t)b * 8 + tid] = (float)freeLen;
  }
  if (tid == 0) {
    feat[(size_t)b * 8 + 3] = (float)s_rot[0];
    feat[(size_t)b * 8 + 4] = (float)s_rot[1];
    feat[(size_t)b * 8 + 5] = 0.0f;
    feat[(size_t)b * 8 + 6] = 0.0f;
    feat[(size_t)b * 8 + 7] = 0.0f;
  }
}

// ---------------------------------------------------------------------------
// Kernel 2: MLP  out = relu(F @ W1^T + b1) @ W2^T + b2  via V_WMMA_F32_16X16X4_F32.
// One wave (32 threads) handles 16 batch rows.
//
// f32 WMMA VGPR layouts (ISA 7.12.2):
//   A (16x4, MxK):  lanes 0-15: M=lane, v0=K0, v1=K1 ; lanes 16-31: v0=K2, v1=K3
//   B (4x16, KxN):  mirror of A: lanes hold N, lane-half selects K pair
//   C/D (16x16):    lane = N (mod 16), VGPR v = row M = v + 8*(lane>=16)
// ---------------------------------------------------------------------------
static __device__ inline v8f splat8(float v) {
  v8f r;
#pragma unroll
  for (int i = 0; i < 8; ++i) r[i] = v;
  return r;
}

__global__ void snake_mlp_kernel(const float* __restrict__ feat,
                                 const float* __restrict__ w1,   // (20,5)
                                 const float* __restrict__ b1,   // (20)
                                 const float* __restrict__ w2,   // (3,20)
                                 const float* __restrict__ b2,   // (3)
                                 float* __restrict__ out) {      // (B,3)
  const int lane = threadIdx.x;         // 32 lanes, all active
  const int base = blockIdx.x * 16;     // 16 batch rows per wave
  const int M  = lane & 15;             // A-row for this lane
  const int N  = lane & 15;             // C/D-column for this lane
  const int kh = lane >> 4;             // lane-half selects K pair / M offset

  __shared__ float Hs[16][20];          // hidden activations (per-wave block)

  // ---- Layer 1: H = relu(F @ W1^T + b1), 16x20, as two N-tiles ----
  const float* frow = feat + (size_t)(base + M) * 8;
  v2f a0; a0.x = frow[0 + 2 * kh]; a0.y = frow[1 + 2 * kh];  // K = 0..3
  v2f a1; a1.x = frow[4 + 2 * kh]; a1.y = frow[5 + 2 * kh];  // K = 4..7 (5..7 == 0)

  v8f h[2];
#pragma unroll
  for (int t = 0; t < 2; ++t) {
    const int n = t * 16 + N;                               // hidden unit
    const bool nv = (n < 20);
    // B[k][n] = w1[n][k], zero-padded for k>=5 or n>=20
    const int k0a = 2 * kh;
    const int k0b = 4 + 2 * kh;
    v2f bA; bA.x = nv ? w1[n * 5 + k0a] : 0.0f;
            bA.y = nv ? w1[n * 5 + k0a + 1] : 0.0f;
    v2f bB; bB.x = (nv && k0b < 5) ? w1[n * 5 + k0b] : 0.0f;
            bB.y = 0.0f;                                    // k0b+1 >= 5 always
    v8f c = splat8(nv ? b1[n] : 0.0f);
    c = __builtin_amdgcn_wmma_f32_16x16x4_f32(false, a0, false, bA,
                                              (short)0, c, false, false);
    c = __builtin_amdgcn_wmma_f32_16x16x4_f32(false, a1, false, bB,
                                              (short)0, c, false, false);
#pragma unroll
    for (int v = 0; v < 8; ++v) c[v] = fmaxf(c[v], 0.0f);   // ReLU
    h[t] = c;
  }

  // ---- transpose H (C/D layout -> LDS row-major) ----
#pragma unroll
  for (int v = 0; v < 8; ++v) {
    const int row = v + 8 * kh;
    Hs[row][N] = h[0][v];
    if (N < 4) Hs[row][16 + N] = h[1][v];
  }
  __syncthreads();

  // ---- Layer 2: out = H @ W2^T + b2, K = 20 in 5 chunks of 4 ----
  const bool ov = (N < 3);
  v8f o = splat8(ov ? b2[N] : 0.0f);
#pragma unroll
  for (int c5 = 0; c5 < 5; ++c5) {
    const int k = 4 * c5 + 2 * kh;                          // k, k+1 < 20
    v2f a; a.x = Hs[M][k]; a.y = Hs[M][k + 1];
    v2f bb; bb.x = ov ? w2[N * 20 + k] : 0.0f;
            bb.y = ov ? w2[N * 20 + k + 1] : 0.0f;
    o = __builtin_amdgcn_wmma_f32_16x16x4_f32(false, a, false, bb,
                                              (short)0, o, false, false);
  }

  // ---- store: lane (N<3, both halves) writes rows M=v+8*kh ----
  if (ov) {
#pragma unroll
    for (int v = 0; v < 8; ++v) {
      const int row = base + v + 8 * kh;
      out[(size_t)row * 3 + N] = o[v];
    }
  }
}

extern "C" void kernel_launch(void* const* d_in, const int* in_sizes, int n_in,
                              void* d_out, int out_size, void* d_ws, size_t ws_size,
                              hipStream_t stream) {
  const float* x  = (const float*)d_in[0];
  const float* w1 = (const float*)d_in[1];
  const float* b1 = (const float*)d_in[2];
  const float* w2 = (const float*)d_in[3];
  const float* b2 = (const float*)d_in[4];
  float* outp = (float*)d_out;
  float* featws = (float*)d_ws;               // B * 8 floats

  const int B = in_sizes[0] / 3072;           // 8192

  snake_feat_kernel<<<B, 256, 0, stream>>>(x, featws);
  snake_mlp_kernel<<<B / 16, 32, 0, stream>>>(featws, w1, b1, w2, b2, outp);
}